// UGAT_13666585936156
// MI455X (gfx1250) — compile-verified
//
#include <hip/hip_runtime.h>
#include <math.h>

// ---------------------------------------------------------------------------
// UGAT: 3-layer GATConv (H=1) + link-prediction scores, MI455X (gfx1250).
// GEMMs use V_WMMA_F32_16X16X4_F32 (wave32, 16x16 tile per wave).
// Edge softmax uses ordered-uint atomic max + f32 atomic adds (L2-resident).
// ---------------------------------------------------------------------------

typedef __attribute__((ext_vector_type(2))) float v2f;
typedef __attribute__((ext_vector_type(8))) float v8f;

#define N_NODES 50000
#define N_EDGES 800000
#define NODE_PAD 50048   // padded per-node scalar buffer stride (floats)

// --- order-preserving float<->uint encoding for atomic max over floats -----
__device__ __forceinline__ unsigned enc_f32(float f) {
  unsigned b = __float_as_uint(f);
  return (b & 0x80000000u) ? ~b : (b | 0x80000000u);
}
__device__ __forceinline__ float dec_f32(unsigned k) {
  unsigned b = (k & 0x80000000u) ? (k ^ 0x80000000u) : ~k;
  return __uint_as_float(b);
}
#define ENC_NEG_INF 0x007FFFFFu   // enc_f32(-inf)

// ---------------------------------------------------------------------------
// fp32 WMMA GEMM: C[M,N] = A[M,K] * B[K,N].  One wave per 16x16 C tile.
// grid = (M/16, N/16), block = 32.  M=50000=3125*16, K,N multiples of 4/16.
// A 16x4 layout: lanes 0-15 -> K=k0,k0+1 ; lanes 16-31 -> K=k0+2,k0+3 (M=lane&15)
// B 4x16 layout: symmetric with N=lane&15.
// C/D: VGPR r -> row r + 8*(lane>>4), col lane&15.
// ---------------------------------------------------------------------------
__global__ __launch_bounds__(32)
void ugat_gemm_wmma(const float* __restrict__ A, const float* __restrict__ B,
                    float* __restrict__ C, int K, int N) {
  const int lane = threadIdx.x;
  const int half = lane >> 4;
  const int lm   = lane & 15;
  const int row0 = blockIdx.x * 16;
  const int col0 = blockIdx.y * 16;

  const float* __restrict__ arow = A + (size_t)(row0 + lm) * (size_t)K;
  const float* __restrict__ bcol = B + (size_t)(col0 + lm);

  v8f acc = {};
  #pragma unroll 4
  for (int k0 = 0; k0 < K; k0 += 4) {
    const int ka = k0 + 2 * half;              // even -> 8B-aligned float2 load
    v2f a = *(const v2f*)(arow + ka);
    v2f b;
    b.x = bcol[(size_t)ka * (size_t)N];
    b.y = bcol[(size_t)(ka + 1) * (size_t)N];
    acc = __builtin_amdgcn_wmma_f32_16x16x4_f32(
        /*neg_a=*/false, a, /*neg_b=*/false, b,
        /*c_mod=*/(short)0, acc, /*reuse_a=*/false, /*reuse_b=*/false);
  }

  float* crow = C + (size_t)(row0 + half * 8) * (size_t)N + (size_t)(col0 + lm);
  #pragma unroll
  for (int r = 0; r < 8; ++r) crow[(size_t)r * (size_t)N] = acc[r];
}

// --- fills ------------------------------------------------------------------
__global__ void ugat_fill_f32(float* __restrict__ p, float v, int n) {
  int i = blockIdx.x * blockDim.x + threadIdx.x;
  if (i < n) p[i] = v;
}
__global__ void ugat_fill_u32(unsigned* __restrict__ p, unsigned v, int n) {
  int i = blockIdx.x * blockDim.x + threadIdx.x;
  if (i < n) p[i] = v;
}

// --- per-node attention logits: el = <feat, al>, er = <feat, ar> (wave/node)
__global__ __launch_bounds__(256)
void ugat_node_logits(const float* __restrict__ feat,
                      const float* __restrict__ al, const float* __restrict__ ar,
                      float* __restrict__ el, float* __restrict__ er,
                      int n, int F) {
  int node = blockIdx.x * (blockDim.x >> 5) + (threadIdx.x >> 5);
  int lane = threadIdx.x & 31;
  if (node >= n) return;
  float sl = 0.f, sr = 0.f;
  const float* frow = feat + (size_t)node * (size_t)F;
  for (int f = lane; f < F; f += 32) {
    float v = frow[f];
    sl += v * al[f];
    sr += v * ar[f];
  }
  #pragma unroll
  for (int off = 16; off > 0; off >>= 1) {
    sl += __shfl_down(sl, off, 32);
    sr += __shfl_down(sr, off, 32);
  }
  if (lane == 0) { el[node] = sl; er[node] = sr; }
}

// --- edge pass 1: e = leaky_relu(el[src]+er[dst]); segment max over dst -----
__global__ __launch_bounds__(256)
void ugat_edge_max(const float* __restrict__ el, const float* __restrict__ er,
                   const int* __restrict__ src, const int* __restrict__ dst,
                   float* __restrict__ eedge, unsigned* __restrict__ emax_bits,
                   int E) {
  int i = blockIdx.x * blockDim.x + threadIdx.x;
  if (i >= E) return;
  float v = el[src[i]] + er[dst[i]];
  float e = (v > 0.f) ? v : 0.2f * v;          // leaky_relu, slope 0.2
  eedge[i] = e;
  atomicMax(&emax_bits[dst[i]], enc_f32(e));
}

// --- decode segment max bits back to float (in place) -----------------------
__global__ void ugat_decode_max(float* __restrict__ emax, int n) {
  int i = blockIdx.x * blockDim.x + threadIdx.x;
  if (i < n) {
    unsigned k = __float_as_uint(emax[i]);
    emax[i] = dec_f32(k);
  }
}

// --- edge pass 2: ex = exp(e - emax[dst]); segment sum over dst -------------
__global__ __launch_bounds__(256)
void ugat_edge_exp(float* __restrict__ eedge, const float* __restrict__ emax,
                   float* __restrict__ denom, const int* __restrict__ dst,
                   int E) {
  int i = blockIdx.x * blockDim.x + threadIdx.x;
  if (i >= E) return;
  int d = dst[i];
  float ex = expf(eedge[i] - emax[d]);
  eedge[i] = ex;
  atomicAdd(&denom[d], ex);
}

// --- edge pass 3: agg[dst] += feat[src] * (ex / denom[dst])  (wave/edge) ----
__global__ __launch_bounds__(256)
void ugat_scatter(const float* __restrict__ feat, const float* __restrict__ eedge,
                  const float* __restrict__ denom,
                  const int* __restrict__ src, const int* __restrict__ dst,
                  float* __restrict__ agg, int E, int F) {
  int edge = blockIdx.x * (blockDim.x >> 5) + (threadIdx.x >> 5);
  int lane = threadIdx.x & 31;
  if (edge >= E) return;
  int s = src[edge], d = dst[edge];
  float alpha = eedge[edge] / denom[d];
  const float* frow = feat + (size_t)s * (size_t)F;
  float* arow = agg + (size_t)d * (size_t)F;
  for (int f = lane; f < F; f += 32)
    atomicAdd(&arow[f], frow[f] * alpha);
}

// --- epilogue: out = (agg + b) with optional relu ---------------------------
__global__ void ugat_epilogue(const float* __restrict__ agg,
                              const float* __restrict__ bias,
                              float* __restrict__ out, int n, int F, int relu) {
  int i = blockIdx.x * blockDim.x + threadIdx.x;
  if (i >= n * F) return;
  float v = agg[i] + bias[i % F];
  if (relu) v = fmaxf(v, 0.f);
  out[i] = v;
}

// --- link predictor: score = [h[a] ; h[b]] @ lp_W + lp_b  (wave/edge, F=64) -
__global__ __launch_bounds__(256)
void ugat_score(const float* __restrict__ h, const float* __restrict__ lpW,
                const float* __restrict__ lpb,
                const int* __restrict__ ia, const int* __restrict__ ib,
                float* __restrict__ out, int E) {
  int edge = blockIdx.x * (blockDim.x >> 5) + (threadIdx.x >> 5);
  int lane = threadIdx.x & 31;
  if (edge >= E) return;
  const float* ha = h + (size_t)ia[edge] * 64;
  const float* hb = h + (size_t)ib[edge] * 64;
  float s = ha[lane]      * lpW[lane]
          + ha[lane + 32] * lpW[lane + 32]
          + hb[lane]      * lpW[64 + lane]
          + hb[lane + 32] * lpW[96 + lane];
  #pragma unroll
  for (int off = 16; off > 0; off >>= 1) s += __shfl_down(s, off, 32);
  if (lane == 0) out[edge] = s + lpb[0];
}

// ---------------------------------------------------------------------------
extern "C" void kernel_launch(void* const* d_in, const int* in_sizes, int n_in,
                              void* d_out, int out_size, void* d_ws, size_t ws_size,
                              hipStream_t stream) {
  (void)in_sizes; (void)n_in; (void)out_size; (void)ws_size;

  const float* x   = (const float*)d_in[0];
  const float* W1  = (const float*)d_in[1];
  const float* al1 = (const float*)d_in[2];
  const float* ar1 = (const float*)d_in[3];
  const float* b1  = (const float*)d_in[4];
  const float* W2  = (const float*)d_in[5];
  const float* al2 = (const float*)d_in[6];
  const float* ar2 = (const float*)d_in[7];
  const float* b2  = (const float*)d_in[8];
  const float* W3  = (const float*)d_in[9];
  const float* al3 = (const float*)d_in[10];
  const float* ar3 = (const float*)d_in[11];
  const float* b3  = (const float*)d_in[12];
  const float* lpW = (const float*)d_in[13];
  const float* lpb = (const float*)d_in[14];
  const int* src     = (const int*)d_in[15];
  const int* dst     = (const int*)d_in[16];
  const int* neg_src = (const int*)d_in[17];
  const int* neg_dst = (const int*)d_in[18];

  // workspace layout (floats): ~80.8 MB total — L2-resident on MI455X (192MB)
  float* ws   = (float*)d_ws;
  float* feat = ws;                                   // [50000,128] projection
  float* hbuf = feat + (size_t)N_NODES * 128;         // [50000,128] layer io
  float* agg  = hbuf + (size_t)N_NODES * 128;         // [50000,128] aggregation
  float* el   = agg  + (size_t)N_NODES * 128;         // [50000]
  float* er   = el   + NODE_PAD;
  float* emx  = er   + NODE_PAD;                      // uint bits then float
  float* den  = emx  + NODE_PAD;
  float* ee   = den  + NODE_PAD;                      // [800000] per-edge e/ex

  const int TB = 256;
  const dim3 blk(TB);
  const int gNodesWave = (N_NODES + (TB / 32) - 1) / (TB / 32);   // wave/node
  const int gNodes     = (N_NODES + TB - 1) / TB;
  const int gEdges     = (N_EDGES + TB - 1) / TB;
  const int gEdgesWave = (N_EDGES + (TB / 32) - 1) / (TB / 32);   // wave/edge

  float* h_out = (float*)d_out;                 // [50000,64]
  float* pos   = h_out + (size_t)N_NODES * 64;  // [800000]
  float* neg   = pos + N_EDGES;                 // [800000]

  // one GAT layer: hin[K] -> hout[F];  hout = act(segsum(alpha * (hin@W)) + b)
  auto gat_layer = [&](const float* hin, int K, const float* W,
                       const float* al, const float* ar, const float* bias,
                       int F, int relu, float* hout) {
    ugat_gemm_wmma<<<dim3(N_NODES / 16, F / 16), dim3(32), 0, stream>>>(hin, W, feat, K, F);
    ugat_node_logits<<<gNodesWave, blk, 0, stream>>>(feat, al, ar, el, er, N_NODES, F);
    ugat_fill_u32<<<gNodes, blk, 0, stream>>>((unsigned*)emx, ENC_NEG_INF, N_NODES);
    ugat_fill_f32<<<gNodes, blk, 0, stream>>>(den, 0.f, N_NODES);
    ugat_fill_f32<<<(N_NODES * F + TB - 1) / TB, blk, 0, stream>>>(agg, 0.f, N_NODES * F);
    ugat_edge_max<<<gEdges, blk, 0, stream>>>(el, er, src, dst, ee, (unsigned*)emx, N_EDGES);
    ugat_decode_max<<<gNodes, blk, 0, stream>>>(emx, N_NODES);
    ugat_edge_exp<<<gEdges, blk, 0, stream>>>(ee, emx, den, dst, N_EDGES);
    ugat_scatter<<<gEdgesWave, blk, 0, stream>>>(feat, ee, den, src, dst, agg, N_EDGES, F);
    ugat_epilogue<<<(N_NODES * F + TB - 1) / TB, blk, 0, stream>>>(agg, bias, hout, N_NODES, F, relu);
  };

  gat_layer(x,    128, W1, al1, ar1, b1,  64, 1, hbuf);   // 128 -> 64, relu
  gat_layer(hbuf,  64, W2, al2, ar2, b2, 128, 1, hbuf);   //  64 -> 128, relu
  gat_layer(hbuf, 128, W3, al3, ar3, b3,  64, 0, h_out);  // 128 -> 64

  ugat_score<<<gEdgesWave, blk, 0, stream>>>(h_out, lpW, lpb, src, dst, pos, N_EDGES);
  ugat_score<<<gEdgesWave, blk, 0, stream>>>(h_out, lpW, lpb, neg_src, neg_dst, neg, N_EDGES);
}